// HierarchicalJointModelBERTDAN_5128190951509
// MI455X (gfx1250) — compile-verified
//
#include <hip/hip_runtime.h>
#include <math.h>

typedef __attribute__((ext_vector_type(2))) float v2f;
typedef __attribute__((ext_vector_type(8))) float v8f;

#define S_DIM 64
#define H_DIM 256
#define K_DIM 768
#define KC 32
#define NCHUNK (K_DIM / KC)    // 24
#define UFD 20
#define PB_DIM (H_DIM + UFD)   // 276
#define LABELS 9
#define NB 64
#define NR 16

#if __has_builtin(__builtin_amdgcn_global_load_async_to_lds_b128)
#define HAVE_ASYNC_LDS 1
#define A_NBUF 2
#else
#define HAVE_ASYNC_LDS 0
#define A_NBUF 1
#endif

#if HAVE_ASYNC_LDS
typedef int v4i_b __attribute__((vector_size(16)));
typedef __attribute__((address_space(1))) v4i_b* as1_v4i_ptr;
typedef __attribute__((address_space(3))) v4i_b* as3_v4i_ptr;

__device__ __forceinline__ void async_copy16(const float* g, float* l) {
    __builtin_amdgcn_global_load_async_to_lds_b128(
        (as1_v4i_ptr)(void*)g,
        (as3_v4i_ptr)(void*)l,
        /*offset=*/0, /*cpol=*/0);
}
#endif

__device__ __forceinline__ void wait_async2() {
#if __has_builtin(__builtin_amdgcn_s_wait_asynccnt)
    __builtin_amdgcn_s_wait_asynccnt(2);
#else
    asm volatile("s_wait_asynccnt 0x2" ::: "memory");
#endif
}
__device__ __forceinline__ void wait_async0() {
#if __has_builtin(__builtin_amdgcn_s_wait_asynccnt)
    __builtin_amdgcn_s_wait_asynccnt(0);
#else
    asm volatile("s_wait_asynccnt 0x0" ::: "memory");
#endif
}

// ---------------------------------------------------------------------------
// Kernel 1: per-(b,r) fused projection GEMM + sentence-level masked mean.
// grid = B*R = 1024 blocks, 256 threads (8 waves).
// Each wave: 2 N-tiles x 4 M-tiles of 16x16, K-loop of V_WMMA_F32_16X16X4_F32.
// A tile staged via async global->LDS (double buffered); W tile staged through
// software-pipelined registers so global loads stay in flight behind WMMAs.
// ---------------------------------------------------------------------------
__global__ void __launch_bounds__(256)
proj_dan_kernel(const float* __restrict__ inputs,   // [B,R,S,768]
                const float* __restrict__ W_proj,   // [768,256]
                const float* __restrict__ b_proj,   // [256]
                float* __restrict__ ws_rhn,         // [B*R, 256] masked r_hn * r_mask
                float* __restrict__ ws_rmask)       // [B*R]
{
    __shared__ __align__(16) float sA[A_NBUF][S_DIM][KC + 4]; // stride 36 floats
    __shared__ __align__(16) float sW[H_DIM][KC + 4];         // W^T slice: [n][k]
    __shared__ float rowsum[S_DIM];
    __shared__ float colsum[H_DIM];

    const int tid  = threadIdx.x;
    const int wave = tid >> 5;
    const int lane = tid & 31;
    const int lrow = lane & 15;
    const int half = lane >> 4;

    if (tid < S_DIM) rowsum[tid] = 0.0f;
    colsum[tid] = 0.0f;

    const float* Ablk = inputs + (size_t)blockIdx.x * S_DIM * K_DIM;

    v8f acc[4][2];
    #pragma unroll
    for (int mt = 0; mt < 4; ++mt)
        #pragma unroll
        for (int nt = 0; nt < 2; ++nt)
            acc[mt][nt] = (v8f){0,0,0,0,0,0,0,0};

    const int n0base = wave * 32;

    // ---- staging helpers ----
    auto loadW = [&](int k0, float4* w) {
        #pragma unroll
        for (int it = 0; it < 8; ++it) {
            int idx = tid + 256 * it;
            int k   = idx >> 6;              // 64 float4 per k-row
            int n4  = (idx & 63) * 4;
            w[it] = *(const float4*)&W_proj[(size_t)(k0 + k) * H_DIM + n4];
        }
    };
    auto storeW = [&](const float4* w) {
        #pragma unroll
        for (int it = 0; it < 8; ++it) {
            int idx = tid + 256 * it;
            int k   = idx >> 6;
            int n4  = (idx & 63) * 4;
            sW[n4 + 0][k] = w[it].x;
            sW[n4 + 1][k] = w[it].y;
            sW[n4 + 2][k] = w[it].z;
            sW[n4 + 3][k] = w[it].w;
        }
    };
#if HAVE_ASYNC_LDS
    auto issueA = [&](int k0, int buf) {
        #pragma unroll
        for (int it = 0; it < 2; ++it) {
            int idx = tid + 256 * it;
            int s   = idx >> 3;              // 8 float4 per row
            int kq  = (idx & 7) * 4;
            async_copy16(&Ablk[(size_t)s * K_DIM + k0 + kq], &sA[buf][s][kq]);
        }
    };
#else
    auto loadA = [&](int k0, float4* a) {
        #pragma unroll
        for (int it = 0; it < 2; ++it) {
            int idx = tid + 256 * it;
            int s   = idx >> 3;
            int kq  = (idx & 7) * 4;
            a[it] = *(const float4*)&Ablk[(size_t)s * K_DIM + k0 + kq];
        }
    };
    auto storeA = [&](const float4* a) {
        #pragma unroll
        for (int it = 0; it < 2; ++it) {
            int idx = tid + 256 * it;
            int s   = idx >> 3;
            int kq  = (idx & 7) * 4;
            *(float4*)&sA[0][s][kq] = a[it];
        }
    };
#endif

    // ---- prologue: start chunk 0 ----
    float4 wreg[8];
#if HAVE_ASYNC_LDS
    issueA(0, 0);
#else
    float4 areg[2];
    loadA(0, areg);
#endif
    loadW(0, wreg);

    for (int c = 0; c < NCHUNK; ++c) {
        const int buf = c & (A_NBUF - 1);

        // commit current chunk's W (and A in fallback) to LDS
        storeW(wreg);
#if !HAVE_ASYNC_LDS
        storeA(areg);
#endif
        // start next chunk's transfers (stay in flight behind the WMMAs)
        if (c + 1 < NCHUNK) {
            const int k0n = (c + 1) * KC;
#if HAVE_ASYNC_LDS
            issueA(k0n, buf ^ 1);
#else
            loadA(k0n, areg);
#endif
            loadW(k0n, wreg);
        }
#if HAVE_ASYNC_LDS
        if (c + 1 < NCHUNK) wait_async2();   // chunk c's asyncs retired
        else                wait_async0();
#endif
        __syncthreads();

        // ---- WMMA K loop over this chunk ----
        #pragma unroll
        for (int kc = 0; kc < KC; kc += 4) {
            const int koff = kc + half * 2;
            v2f a[4], bmat[2];
            #pragma unroll
            for (int mt = 0; mt < 4; ++mt)
                a[mt] = *(const v2f*)&sA[buf][mt * 16 + lrow][koff];
            #pragma unroll
            for (int nt = 0; nt < 2; ++nt)
                bmat[nt] = *(const v2f*)&sW[n0base + nt * 16 + lrow][koff];
            #pragma unroll
            for (int mt = 0; mt < 4; ++mt)
                #pragma unroll
                for (int nt = 0; nt < 2; ++nt)
                    acc[mt][nt] = __builtin_amdgcn_wmma_f32_16x16x4_f32(
                        false, a[mt], false, bmat[nt],
                        (short)0, acc[mt][nt], false, false);
        }
        __syncthreads();
    }

    // ---- epilogue: add bias ----
    #pragma unroll
    for (int nt = 0; nt < 2; ++nt) {
        float bp = b_proj[n0base + nt * 16 + lrow];
        #pragma unroll
        for (int mt = 0; mt < 4; ++mt)
            #pragma unroll
            for (int v = 0; v < 8; ++v)
                acc[mt][nt][v] += bp;
    }

    // ---- per-row sums (for s_mask) ----
    #pragma unroll
    for (int mt = 0; mt < 4; ++mt)
        #pragma unroll
        for (int v = 0; v < 8; ++v) {
            int row = mt * 16 + half * 8 + v;
            atomicAdd(&rowsum[row], acc[mt][0][v] + acc[mt][1][v]);
        }
    __syncthreads();

    // ---- den (sum of s_mask) and total (for r_mask), redundantly per thread ----
    float den = 0.0f, total = 0.0f;
    #pragma unroll 8
    for (int s = 0; s < S_DIM; ++s) {
        float rs = rowsum[s];
        den   += (rs != 0.0f) ? 1.0f : 0.0f;
        total += rs;
    }

    // ---- masked column sums ----
    #pragma unroll
    for (int nt = 0; nt < 2; ++nt) {
        float part = 0.0f;
        #pragma unroll
        for (int mt = 0; mt < 4; ++mt)
            #pragma unroll
            for (int v = 0; v < 8; ++v) {
                int row = mt * 16 + half * 8 + v;
                float m = (rowsum[row] != 0.0f) ? 1.0f : 0.0f;
                part += m * acc[mt][nt][v];
            }
        atomicAdd(&colsum[n0base + nt * 16 + lrow], part);
    }
    __syncthreads();

    // ---- write masked review vector ----
    float rmask = (total != 0.0f) ? 1.0f : 0.0f;
    float denr  = fmaxf(den, 1e-9f);
    ws_rhn[(size_t)blockIdx.x * H_DIM + tid] = rmask * colsum[tid] / denr;
    if (tid == 0) ws_rmask[blockIdx.x] = rmask;
}

// ---------------------------------------------------------------------------
// Kernel 2: review-level DAN, user-feature normalize, SELU heads.
// single block, 256 threads.
// ---------------------------------------------------------------------------
__device__ __forceinline__ float selu_f(float x) {
    const float scale = 1.0507009873554805f;
    const float alpha = 1.6732632423543772f;
    return scale * (x > 0.0f ? x : alpha * (expf(x) - 1.0f));
}

__global__ void __launch_bounds__(256)
tail_kernel(const float* __restrict__ ws_rhn,     // [B*R,256]
            const float* __restrict__ ws_rmask,   // [B*R]
            const float* __restrict__ user_feats, // [B,20]
            const float* __restrict__ W_rff,      // [276,1]
            const float* __restrict__ b_rff,      // [1]
            const float* __restrict__ W_pff,      // [276,9]
            const float* __restrict__ b_pff,      // [9]
            const float* __restrict__ user_w,     // [20]
            float* __restrict__ out)              // [9 + 64]
{
    __shared__ float pb[PB_DIM];
    __shared__ float hn_acc[PB_DIM];
    __shared__ float red[256];

    const int tid = threadIdx.x;
    for (int j = tid; j < PB_DIM; j += 256) hn_acc[j] = 0.0f;

    for (int b = 0; b < NB; ++b) {
        float den = 0.0f;
        #pragma unroll
        for (int r = 0; r < NR; ++r) den += ws_rmask[b * NR + r];
        float denr = fmaxf(den, 1e-9f);

        float num = 0.0f;
        #pragma unroll
        for (int r = 0; r < NR; ++r)
            num += ws_rhn[(size_t)(b * NR + r) * H_DIM + tid];
        pb[tid] = num / denr;

        if (tid < UFD) {
            float ss = 0.0f;
            #pragma unroll
            for (int j = 0; j < UFD; ++j) {
                float v = user_feats[b * UFD + j];
                ss += v * v;
            }
            float nrm = fmaxf(sqrtf(ss), 1e-12f);
            pb[H_DIM + tid] = (user_feats[b * UFD + tid] / nrm) * user_w[tid];
        }
        __syncthreads();

        float part = pb[tid] * W_rff[tid];
        if (tid < UFD) part += pb[H_DIM + tid] * W_rff[H_DIM + tid];
        red[tid] = part;
        __syncthreads();
        for (int off = 128; off > 0; off >>= 1) {
            if (tid < off) red[tid] += red[tid + off];
            __syncthreads();
        }
        if (tid == 0) out[LABELS + b] = selu_f(red[0] + b_rff[0]);

        for (int j = tid; j < PB_DIM; j += 256) hn_acc[j] += pb[j];
        __syncthreads();
    }

    if (tid < LABELS) {
        float s = b_pff[tid];
        for (int j = 0; j < PB_DIM; ++j)
            s += (hn_acc[j] * (1.0f / (float)NB)) * W_pff[j * LABELS + tid];
        out[tid] = selu_f(s);
    }
}

// ---------------------------------------------------------------------------
extern "C" void kernel_launch(void* const* d_in, const int* in_sizes, int n_in,
                              void* d_out, int out_size, void* d_ws, size_t ws_size,
                              hipStream_t stream) {
    const float* inputs     = (const float*)d_in[0];
    const float* user_feats = (const float*)d_in[1];
    const float* W_proj     = (const float*)d_in[2];
    const float* b_proj     = (const float*)d_in[3];
    const float* W_rff      = (const float*)d_in[4];
    const float* b_rff      = (const float*)d_in[5];
    const float* W_pff      = (const float*)d_in[6];
    const float* b_pff      = (const float*)d_in[7];
    const float* user_w     = (const float*)d_in[8];
    float* out = (float*)d_out;

    float* ws_rhn   = (float*)d_ws;                     // 64*16*256 floats
    float* ws_rmask = ws_rhn + (size_t)NB * NR * H_DIM; // 1024 floats

    proj_dan_kernel<<<NB * NR, 256, 0, stream>>>(inputs, W_proj, b_proj,
                                                 ws_rhn, ws_rmask);
    tail_kernel<<<1, 256, 0, stream>>>(ws_rhn, ws_rmask, user_feats,
                                       W_rff, b_rff, W_pff, b_pff, user_w, out);
}